// ScalarDotProductCriticNetwork_V1_32590211842699
// MI455X (gfx1250) — compile-verified
//
#include <hip/hip_runtime.h>

// MI455X / gfx1250, wave32. fp32 WMMA path: V_WMMA_F32_16X16X4_F32.
typedef __attribute__((ext_vector_type(2))) float v2f;
typedef __attribute__((ext_vector_type(8))) float v8f;

#define NB   512
#define NAGT 32
#define OBSD 128
#define NACT 16
#define OAD  144
#define DEMB 64

// ---- LDS layout (floats) ----
// B operands (K-major) use stride 80 (== 16 mod 64): conflict-free dword loads.
// A operands (row-major) use stride == 4 mod 64: conflict-free b64 loads.
constexpr int WS     = 80;
constexpr int O_WOBS = 0;                 // [128][80]  W_obs^T
constexpr int O_WOA  = O_WOBS + 128*WS;   // [144][80]  W_oa^T
constexpr int O_WKEY = O_WOA  + 144*WS;   // [64][80]
constexpr int O_WQRY = O_WKEY + 64*WS;
constexpr int O_WV   = O_WQRY + 64*WS;
constexpr int O_WF1  = O_WV   + 64*WS;
constexpr int O_WF2  = O_WF1  + 64*WS;    // [64][20]
constexpr int O_BOBS = O_WF2  + 64*20;    // [64]
constexpr int O_BOA  = O_BOBS + 64;       // [64]
constexpr int O_S    = O_BOA  + 64;       // [32][132] states
constexpr int O_ACT  = O_S    + 32*132;   // [32][20]
constexpr int O_POL  = O_ACT  + 32*20;    // [32][20]
constexpr int O_E    = O_POL  + 32*20;    // [32][68] states_embed
constexpr int O_KEY  = O_E    + 32*68;    // [32][68]
constexpr int O_QT   = O_KEY  + 32*68;    // [64][40] query^T (B operand of score)
constexpr int O_EACT = O_QT   + 64*40;    // [32][68]
constexpr int O_EPOL = O_EACT + 32*68;    // [32][68]
constexpr int O_AACT = O_EPOL + 32*68;    // [32][80] attn_act (B operand of nf)
constexpr int O_APOL = O_AACT + 32*WS;    // [32][68]
constexpr int O_WGT  = O_APOL + 32*68;    // [32][36] score -> softmax weight
constexpr int O_NF   = O_WGT  + 32*36;    // [32][68]
constexpr int O_H    = O_NF   + 32*68;    // [32][68]
constexpr int SMEM_FLOATS = O_H + 32*68;  // 70656 floats = 276 KB < 320 KB/WGP

// 16x16 (MxN) f32 WMMA tile, accumulating KLEN/4 steps of K=4.
// A: row-major [16+][sA] at base; B: K-major [KLEN][sB] at base (col offset applied).
// Lane mapping (ISA 7.12.2): A: row=lane&15, k=k0+2*(lane>>4)+v ; B: n=lane&15, same k.
template <int KLEN>
__device__ __forceinline__ v8f wmma_acc(const float* A, int sA,
                                        const float* B, int sB,
                                        v8f c, int r, int hi) {
#pragma unroll
  for (int k0 = 0; k0 < KLEN; k0 += 4) {
    const int ka = k0 + 2*hi;
    v2f a, bb;
    a.x  = A[r*sA + ka];
    a.y  = A[r*sA + ka + 1];
    bb.x = B[ka*sB + r];
    bb.y = B[(ka + 1)*sB + r];
    c = __builtin_amdgcn_wmma_f32_16x16x4_f32(false, a, false, bb,
                                              (short)0, c, false, false);
  }
  return c;
}

__global__ __launch_bounds__(256, 1)
void critic_attn_kernel(const float* __restrict__ states,
                        const float* __restrict__ policies,
                        const float* __restrict__ actions,
                        const float* __restrict__ W_obs, const float* __restrict__ b_obs,
                        const float* __restrict__ W_oa,  const float* __restrict__ b_oa,
                        const float* __restrict__ W_key, const float* __restrict__ W_qry,
                        const float* __restrict__ W_v,   const float* __restrict__ W_f1,
                        const float* __restrict__ W_f2,
                        float* __restrict__ out_val,     // [512,32,16]
                        float* __restrict__ out_wgt) {   // [512,32,32]
  extern __shared__ float sm[];
  const int tid = threadIdx.x;
  const int b   = blockIdx.x;

  // ---- stage weights transposed (K-major) + inputs into LDS ----
  auto stageT = [&](int dstOff, const float* src, int N, int K, int sD) {
    for (int i = tid; i < N * K; i += 256) {     // coalesced global read
      int n = i / K, k = i - n * K;
      sm[dstOff + k * sD + n] = src[i];
    }
  };
  {
    const float* sb = states + (size_t)b * NAGT * OBSD;
    for (int i = tid; i < NAGT * OBSD; i += 256)
      sm[O_S + (i >> 7) * 132 + (i & 127)] = sb[i];
    const float* ab = actions  + (size_t)b * NAGT * NACT;
    const float* pb = policies + (size_t)b * NAGT * NACT;
    for (int i = tid; i < NAGT * NACT; i += 256) {
      int rr = i >> 4, cc = i & 15;
      sm[O_ACT + rr * 20 + cc] = ab[i];
      sm[O_POL + rr * 20 + cc] = pb[i];
    }
    stageT(O_WOBS, W_obs, 64, 128, WS);
    stageT(O_WOA,  W_oa,  64, 144, WS);
    stageT(O_WKEY, W_key, 64, 64,  WS);
    stageT(O_WQRY, W_qry, 64, 64,  WS);
    stageT(O_WV,   W_v,   64, 64,  WS);
    stageT(O_WF1,  W_f1,  64, 64,  WS);
    stageT(O_WF2,  W_f2,  16, 64,  20);
    if (tid < 64) { sm[O_BOBS + tid] = b_obs[tid]; sm[O_BOA + tid] = b_oa[tid]; }
  }
  __syncthreads();

  const int w    = tid >> 5;
  const int lane = tid & 31;
  const int r    = lane & 15;
  const int hi   = lane >> 4;
  const int mt = w >> 2, nt = w & 3;   // 2x4 tile grid for 32x64 outputs
  const int mB = mt * 16, nB = nt * 16;

  // ---- states_embed = relu(S @ Wobs^T + b_obs)   and  emb_act / emb_pol ----
  {
    v8f c = {};
    c = wmma_acc<128>(&sm[O_S + mB * 132], 132, &sm[O_WOBS + nB], WS, c, r, hi);
#pragma unroll
    for (int v = 0; v < 8; ++v) {
      int m = mB + 8 * hi + v, n = nB + r;
      float x = c[v] + sm[O_BOBS + n];
      sm[O_E + m * 68 + n] = x > 0.f ? x : 0.f;
    }
    // shared K=0..127 (states) accumulation, then two K=16 tails
    v8f cc = {};
    cc = wmma_acc<128>(&sm[O_S + mB * 132], 132, &sm[O_WOA + nB], WS, cc, r, hi);
    v8f ca = wmma_acc<16>(&sm[O_ACT + mB * 20], 20, &sm[O_WOA + 128 * WS + nB], WS, cc, r, hi);
    v8f cp = wmma_acc<16>(&sm[O_POL + mB * 20], 20, &sm[O_WOA + 128 * WS + nB], WS, cc, r, hi);
#pragma unroll
    for (int v = 0; v < 8; ++v) {
      int m = mB + 8 * hi + v, n = nB + r;
      float ba = sm[O_BOA + n];
      float xa = ca[v] + ba, xp = cp[v] + ba;
      sm[O_EACT + m * 68 + n] = xa > 0.f ? xa : 0.f;
      sm[O_EPOL + m * 68 + n] = xp > 0.f ? xp : 0.f;
    }
  }
  __syncthreads();

  // ---- key, query (stored transposed), attn_act = tanh(emb@Wv^T), attn_pol ----
  {
    v8f ck = {};
    ck = wmma_acc<64>(&sm[O_E + mB * 68], 68, &sm[O_WKEY + nB], WS, ck, r, hi);
    v8f cq = {};
    cq = wmma_acc<64>(&sm[O_E + mB * 68], 68, &sm[O_WQRY + nB], WS, cq, r, hi);
    v8f ca = {};
    ca = wmma_acc<64>(&sm[O_EACT + mB * 68], 68, &sm[O_WV + nB], WS, ca, r, hi);
    v8f cp = {};
    cp = wmma_acc<64>(&sm[O_EPOL + mB * 68], 68, &sm[O_WV + nB], WS, cp, r, hi);
#pragma unroll
    for (int v = 0; v < 8; ++v) {
      int m = mB + 8 * hi + v, n = nB + r;
      sm[O_KEY + m * 68 + n] = ck[v];
      sm[O_QT + n * 40 + m]  = cq[v];            // transpose: B operand for score
      sm[O_AACT + m * WS + n] = tanhf(ca[v]);
      sm[O_APOL + m * 68 + n] = tanhf(cp[v]);
    }
  }
  __syncthreads();

  // ---- score[j,i] = key_j . query_i / 8  (waves 0..3: 2x2 tiles of 32x32) ----
  if (w < 4) {
    const int smt = w >> 1, snt = w & 1;
    v8f c = {};
    c = wmma_acc<64>(&sm[O_KEY + smt * 16 * 68], 68, &sm[O_QT + snt * 16], 40, c, r, hi);
#pragma unroll
    for (int v = 0; v < 8; ++v) {
      int j = smt * 16 + 8 * hi + v, i = snt * 16 + r;
      sm[O_WGT + j * 36 + i] = c[v] * 0.125f;
    }
  }
  __syncthreads();

  // ---- softmax over i (one row per lane of wave 0), emit weight output ----
  if (tid < 32) {
    const int j = tid;
    float mx = -3.4e38f;
    for (int i = 0; i < 32; ++i) mx = fmaxf(mx, sm[O_WGT + j * 36 + i]);
    float s = 0.f;
    for (int i = 0; i < 32; ++i) s += __expf(sm[O_WGT + j * 36 + i] - mx);
    const float inv = 1.f / s;
    float* wo = out_wgt + ((size_t)b * 32 + j) * 32;
    for (int i = 0; i < 32; ++i) {
      float wv = __expf(sm[O_WGT + j * 36 + i] - mx) * inv;
      sm[O_WGT + j * 36 + i] = wv;
      wo[i] = wv;
    }
  }
  __syncthreads();

  // ---- node_features[j] = (W[j,:]@attn_act + w[j,j]*(attn_pol[j]-attn_act[j]))/32 ----
  {
    v8f c = {};
    c = wmma_acc<32>(&sm[O_WGT + mB * 36], 36, &sm[O_AACT + nB], WS, c, r, hi);
#pragma unroll
    for (int v = 0; v < 8; ++v) {
      int j = mB + 8 * hi + v, d = nB + r;
      float wjj  = sm[O_WGT + j * 36 + j];
      float corr = wjj * (sm[O_APOL + j * 68 + d] - sm[O_AACT + j * WS + d]);
      sm[O_NF + j * 68 + d] = (c[v] + corr) * (1.0f / 32.0f);
    }
  }
  __syncthreads();

  // ---- H = leaky_relu(NF @ Wf1^T) ----
  {
    v8f c = {};
    c = wmma_acc<64>(&sm[O_NF + mB * 68], 68, &sm[O_WF1 + nB], WS, c, r, hi);
#pragma unroll
    for (int v = 0; v < 8; ++v) {
      int m = mB + 8 * hi + v, n = nB + r;
      float x = c[v];
      sm[O_H + m * 68 + n] = x > 0.f ? x : 0.01f * x;
    }
  }
  __syncthreads();

  // ---- value = H @ Wf2^T  (32x16: waves 0..1) ----
  if (w < 2) {
    v8f c = {};
    c = wmma_acc<64>(&sm[O_H + w * 16 * 68], 68, &sm[O_WF2], 20, c, r, hi);
    float* vo = out_val + (size_t)b * 32 * 16;
#pragma unroll
    for (int v = 0; v < 8; ++v) {
      int m = w * 16 + 8 * hi + v;
      vo[m * 16 + r] = c[v];
    }
  }
}

extern "C" void kernel_launch(void* const* d_in, const int* in_sizes, int n_in,
                              void* d_out, int out_size, void* d_ws, size_t ws_size,
                              hipStream_t stream) {
  const float* states   = (const float*)d_in[0];
  const float* policies = (const float*)d_in[1];
  const float* actions  = (const float*)d_in[2];
  const float* W_obs    = (const float*)d_in[3];
  const float* b_obs    = (const float*)d_in[4];
  const float* W_oa     = (const float*)d_in[5];
  const float* b_oa     = (const float*)d_in[6];
  const float* W_key    = (const float*)d_in[7];
  const float* W_qry    = (const float*)d_in[8];
  const float* W_v      = (const float*)d_in[9];
  const float* W_f1     = (const float*)d_in[10];
  const float* W_f2     = (const float*)d_in[11];
  float* out_val = (float*)d_out;                  // [512,32,16] first
  float* out_wgt = out_val + (size_t)NB * NAGT * NACT;  // then [512,32,32]

  const size_t smemBytes = (size_t)SMEM_FLOATS * sizeof(float);
  critic_attn_kernel<<<NB, 256, smemBytes, stream>>>(
      states, policies, actions, W_obs, b_obs, W_oa, b_oa,
      W_key, W_qry, W_v, W_f1, W_f2, out_val, out_wgt);

  (void)in_sizes; (void)n_in; (void)out_size; (void)d_ws; (void)ws_size;
}